// MemoryN2N_78365973282876
// MI455X (gfx1250) — compile-verified
//
#include <hip/hip_runtime.h>
#include <math.h>

#define NPIX 2304      // 48*48
#define CDIM 256
#define KDIM 2048
#define TOPK 204
#define OUT_CH 516     // 256 + 256 + 4
#define SCORE_OFF (OUT_CH * NPIX)

typedef __attribute__((ext_vector_type(2))) float v2f;
typedef __attribute__((ext_vector_type(8))) float v8f;

__device__ __forceinline__ float warpMax(float v) {
  #pragma unroll
  for (int m = 16; m; m >>= 1) v = fmaxf(v, __shfl_xor(v, m, 32));
  return v;
}
__device__ __forceinline__ float warpSum(float v) {
  #pragma unroll
  for (int m = 16; m; m >>= 1) v += __shfl_xor(v, m, 32);
  return v;
}
__device__ __forceinline__ int warpSumI(int v) {
  #pragma unroll
  for (int m = 16; m; m >>= 1) v += __shfl_xor(v, m, 32);
  return v;
}

// ---------------------------------------------------------------------------
// Kernel 1: L2-normalize feat_units rows (2048 x 256) -> fn, and also emit the
// raw transpose featT (256 x 2048) so GEMM2's B-fragments are contiguous.
// One wave per row.
// ---------------------------------------------------------------------------
__global__ void __launch_bounds__(256)
feat_norm_kernel(const float* __restrict__ feat,
                 float* __restrict__ fn,
                 float* __restrict__ featT) {
  const int wid  = threadIdx.x >> 5;
  const int lane = threadIdx.x & 31;
  const int row  = blockIdx.x * 8 + wid;           // 256 blocks * 8 waves = 2048
  const float* src = feat + (size_t)row * CDIM;
  float v[8], ss = 0.f;
  #pragma unroll
  for (int j = 0; j < 8; ++j) { v[j] = src[lane + j * 32]; ss += v[j] * v[j]; }
  ss = warpSum(ss);
  const float inv = 1.f / fmaxf(sqrtf(ss), 1e-12f);
  float* dst = fn + (size_t)row * CDIM;
  #pragma unroll
  for (int j = 0; j < 8; ++j) {
    const int c = lane + j * 32;
    dst[c] = v[j] * inv;
    featT[(size_t)c * KDIM + row] = v[j];          // raw transpose
  }
}

// ---------------------------------------------------------------------------
// Kernel 2: transpose x (C,H,W)->(pixel,c), L2-normalize pixels, and copy raw
// x into out channels [0,256). One thread per pixel.
// ---------------------------------------------------------------------------
__global__ void __launch_bounds__(256)
x_prep_kernel(const float* __restrict__ x,
              float* __restrict__ xn,
              float* __restrict__ out) {
  const int p = blockIdx.x * 256 + threadIdx.x;    // 9 blocks * 256 = 2304
  float ss = 0.f;
  for (int c = 0; c < CDIM; ++c) {
    const float v = x[(size_t)c * NPIX + p];       // coalesced over p
    ss += v * v;
    out[(size_t)c * NPIX + p] = v;                 // pass-through copy
  }
  const float inv = 1.f / fmaxf(sqrtf(ss), 1e-12f);
  float* dst = xn + (size_t)p * CDIM;
  for (int c = 0; c < CDIM; ++c) dst[c] = x[(size_t)c * NPIX + p] * inv;
}

// ---------------------------------------------------------------------------
// Kernel 3: score = xn (2304x256) @ fn^T via V_WMMA_F32_16X16X4_F32.
// 64x64 register-blocked tile per wave: acc[4][4], 8 float2 loads feed
// 16 wmma per k-step (0.5 loads/wmma). 36 x 32 tiles -> 1152 waves.
// __launch_bounds__(256,1): allow up to 256 VGPRs so the 128 accumulator
// VGPRs live in registers (round 2 showed scratch spills at the 128-VGPR cap).
// ---------------------------------------------------------------------------
__global__ void __launch_bounds__(256, 1)
score_gemm_kernel(const float* __restrict__ xn,
                  const float* __restrict__ fn,
                  float* __restrict__ score) {
  const int wid  = threadIdx.x >> 5;
  const int lane = threadIdx.x & 31;
  const int t    = blockIdx.x * 8 + wid;           // 144 blocks -> 1152 tiles
  const int tm   = t >> 5;                         // 0..35  (M blocks of 64)
  const int tn   = t & 31;                         // 0..31  (N blocks of 64)
  const int half = lane >> 4;                      // 0: K {k,k+1}, 1: K {k+2,k+3}
  const int l16  = lane & 15;

  const float* abase = xn + (size_t)(tm * 64 + l16) * CDIM + half * 2;
  const float* bbase = fn + (size_t)(tn * 64 + l16) * CDIM + half * 2;

  v8f acc[4][4] = {};
  #pragma unroll 2
  for (int k = 0; k < CDIM; k += 4) {
    v2f a[4], b[4];
    #pragma unroll
    for (int i = 0; i < 4; ++i) a[i] = *(const v2f*)(abase + (size_t)i * 16 * CDIM + k);
    #pragma unroll
    for (int j = 0; j < 4; ++j) b[j] = *(const v2f*)(bbase + (size_t)j * 16 * CDIM + k);
    #pragma unroll
    for (int i = 0; i < 4; ++i)
      #pragma unroll
      for (int j = 0; j < 4; ++j)
        acc[i][j] = __builtin_amdgcn_wmma_f32_16x16x4_f32(false, a[i], false, b[j],
                                                          (short)0, acc[i][j],
                                                          false, false);
  }
  // C/D layout: VGPR r -> M = sub_m + half*8 + r, N = sub_n + l16
  #pragma unroll
  for (int i = 0; i < 4; ++i) {
    const int mbase = tm * 64 + i * 16 + half * 8;
    #pragma unroll
    for (int j = 0; j < 4; ++j) {
      const int n = tn * 64 + j * 16 + l16;
      #pragma unroll
      for (int r = 0; r < 8; ++r)
        score[(size_t)(mbase + r) * KDIM + n] = acc[i][j][r];
    }
  }
}

// ---------------------------------------------------------------------------
// Kernel 4: per-row softmax stats + exact top-204 threshold (bitwise binary
// search on order-preserving keys) -> masked weight matrix W' and out_y.
// One 256-thread block per row; each thread owns 8 strided columns.
// ---------------------------------------------------------------------------
__global__ void __launch_bounds__(256)
row_topk_kernel(const float* __restrict__ score,
                const float* __restrict__ label,
                float* __restrict__ wp,
                float* __restrict__ out) {
  const int p    = blockIdx.x;
  const int tid  = threadIdx.x;
  const int lane = tid & 31;
  const int wid  = tid >> 5;

  __shared__ float shf[8];
  __shared__ int   shi[8];
  __shared__ float shy[8][4];

  const float* row = score + (size_t)p * KDIM;
  float s[8];
  unsigned key[8];
  #pragma unroll
  for (int j = 0; j < 8; ++j) {
    s[j] = row[tid + j * 256];
    const unsigned u = __float_as_uint(s[j]);
    key[j] = u ^ ((u >> 31) ? 0xFFFFFFFFu : 0x80000000u);  // order-preserving
  }

  // ---- block max ----
  float m = s[0];
  #pragma unroll
  for (int j = 1; j < 8; ++j) m = fmaxf(m, s[j]);
  m = warpMax(m);
  if (lane == 0) shf[wid] = m;
  __syncthreads();
  float bmax = shf[0];
  #pragma unroll
  for (int i = 1; i < 8; ++i) bmax = fmaxf(bmax, shf[i]);
  __syncthreads();

  // ---- block sum of exp ----
  float e[8], z = 0.f;
  #pragma unroll
  for (int j = 0; j < 8; ++j) { e[j] = __expf(s[j] - bmax); z += e[j]; }
  z = warpSum(z);
  if (lane == 0) shf[wid] = z;
  __syncthreads();
  float bz = 0.f;
  #pragma unroll
  for (int i = 0; i < 8; ++i) bz += shf[i];
  __syncthreads();
  const float invZ = 1.f / bz;

  // ---- 204th-largest key via 32-step bitwise binary search ----
  unsigned T = 0;
  for (int bit = 31; bit >= 0; --bit) {
    const unsigned cand = T | (1u << bit);
    int cnt = 0;
    #pragma unroll
    for (int j = 0; j < 8; ++j) cnt += (key[j] >= cand);
    cnt = warpSumI(cnt);
    if (lane == 0) shi[wid] = cnt;
    __syncthreads();
    int bc = 0;
    #pragma unroll
    for (int i = 0; i < 8; ++i) bc += shi[i];
    __syncthreads();
    if (bc >= TOPK) T = cand;
  }

  // ---- emit masked softmax weights + accumulate out_y ----
  float ay0 = 0.f, ay1 = 0.f, ay2 = 0.f, ay3 = 0.f;
  #pragma unroll
  for (int j = 0; j < 8; ++j) {
    const int k = tid + j * 256;
    const float w = (key[j] >= T) ? e[j] * invZ : 0.f;
    wp[(size_t)p * KDIM + k] = w;
    if (w != 0.f) {
      const float* lr = label + (size_t)k * 4;
      ay0 += w * lr[0]; ay1 += w * lr[1]; ay2 += w * lr[2]; ay3 += w * lr[3];
    }
  }
  ay0 = warpSum(ay0); ay1 = warpSum(ay1); ay2 = warpSum(ay2); ay3 = warpSum(ay3);
  if (lane == 0) { shy[wid][0] = ay0; shy[wid][1] = ay1; shy[wid][2] = ay2; shy[wid][3] = ay3; }
  __syncthreads();
  if (tid < 4) {
    float a = 0.f;
    #pragma unroll
    for (int i = 0; i < 8; ++i) a += shy[i][tid];
    out[(size_t)(512 + tid) * NPIX + p] = a;     // out_y channels
  }
}

// ---------------------------------------------------------------------------
// Kernel 5: out_x = W' (2304x2048) @ feat (2048x256) via WMMA f32 16x16x4.
// B sourced from featT (256x2048) -> contiguous float2 fragment loads.
// 32x32 register-blocked tile per wave: acc[2][2], 4 loads / 4 wmma per step.
// 72 x 8 tiles -> 576 waves. D written channel-major as float4 pairs.
// ---------------------------------------------------------------------------
__global__ void __launch_bounds__(256, 1)
outx_gemm_kernel(const float* __restrict__ wp,
                 const float* __restrict__ featT,
                 float* __restrict__ out) {
  const int wid  = threadIdx.x >> 5;
  const int lane = threadIdx.x & 31;
  const int t    = blockIdx.x * 8 + wid;           // 72 blocks -> 576 tiles
  const int tm   = t >> 3;                         // 0..71  (M blocks of 32)
  const int tn   = t & 7;                          // 0..7   (N blocks of 32)
  const int half = lane >> 4;
  const int l16  = lane & 15;

  const float* abase = wp    + (size_t)(tm * 32 + l16) * KDIM + half * 2;
  const float* bbase = featT + (size_t)(tn * 32 + l16) * KDIM + half * 2;

  v8f acc[2][2] = {};
  #pragma unroll 2
  for (int k = 0; k < KDIM; k += 4) {
    v2f a[2], b[2];
    #pragma unroll
    for (int i = 0; i < 2; ++i) a[i] = *(const v2f*)(abase + (size_t)i * 16 * KDIM + k);
    #pragma unroll
    for (int j = 0; j < 2; ++j) b[j] = *(const v2f*)(bbase + (size_t)j * 16 * KDIM + k);
    #pragma unroll
    for (int i = 0; i < 2; ++i)
      #pragma unroll
      for (int j = 0; j < 2; ++j)
        acc[i][j] = __builtin_amdgcn_wmma_f32_16x16x4_f32(false, a[i], false, b[j],
                                                          (short)0, acc[i][j],
                                                          false, false);
  }
  #pragma unroll
  for (int i = 0; i < 2; ++i) {
    const int mbase = tm * 32 + i * 16 + half * 8;   // 8 consecutive pixels
    #pragma unroll
    for (int j = 0; j < 2; ++j) {
      const int ch = 256 + tn * 32 + j * 16 + l16;   // out_x channel
      float* dst = out + (size_t)ch * NPIX + mbase;
      const v8f c = acc[i][j];
      float4 lo = make_float4(c[0], c[1], c[2], c[3]);
      float4 hi = make_float4(c[4], c[5], c[6], c[7]);
      *(float4*)(dst)     = lo;
      *(float4*)(dst + 4) = hi;
    }
  }
}

// ---------------------------------------------------------------------------
extern "C" void kernel_launch(void* const* d_in, const int* in_sizes, int n_in,
                              void* d_out, int out_size, void* d_ws, size_t ws_size,
                              hipStream_t stream) {
  const float* x     = (const float*)d_in[0];   // (1,256,48,48)
  const float* feat  = (const float*)d_in[1];   // (2048,256)
  const float* label = (const float*)d_in[2];   // (2048,4)
  float* out   = (float*)d_out;                 // out (516*2304) ++ score (2304*2048)
  float* score = out + SCORE_OFF;

  float* xn    = (float*)d_ws;                  // 2304*256
  float* fn    = xn + (size_t)NPIX * CDIM;      // 2048*256 (normalized)
  float* wp    = fn + (size_t)KDIM * CDIM;      // 2304*2048 masked weights
  float* featT = wp + (size_t)NPIX * KDIM;      // 256*2048 (raw transpose)

  feat_norm_kernel<<<KDIM / 8, 256, 0, stream>>>(feat, fn, featT);
  x_prep_kernel<<<NPIX / 256, 256, 0, stream>>>(x, xn, out);
  score_gemm_kernel<<<(36 * 32) / 8, 256, 0, stream>>>(xn, fn, score);
  row_topk_kernel<<<NPIX, 256, 0, stream>>>(score, label, wp, out);
  outx_gemm_kernel<<<(72 * 8) / 8, 256, 0, stream>>>(wp, featT, out);
}